// GraphhMLP_stem_6305011991075
// MI455X (gfx1250) — compile-verified
//
#include <hip/hip_runtime.h>
#include <math.h>

// ---------------------------------------------------------------------------
// GraphMLP stem for MI455X (gfx1250, wave32).
// fp32 end-to-end: K=192 GEMMs on V_WMMA_F32_16X16X4_F32 with a fragment-major
// LDS-staged B panel (B fragments = single ds_load_b64, no repacking);
// layer-0 (K=3) as a trivial FMA kernel; edge aggregation with hardware float
// atomics; GraphNorm via per-graph atomic reductions.
// ---------------------------------------------------------------------------

#define TSTEPS 4
#define NG     16          // number of graphs
#define DCH    192         // processing chunk width (d4 = 192; 768 = 4 chunks)
#define KDIM   192         // GEMM K (layers 1 and 2)
#define EPSV   1e-5f

typedef __attribute__((ext_vector_type(2))) float v2f;
typedef __attribute__((ext_vector_type(8))) float v8f;

__device__ __forceinline__ void atomAdd(float* p, float v) {
  // lowers to global_atomic_add_f32 on gfx1250 (no CAS loop)
  unsafeAtomicAdd(p, v);
}

// ---------------------------------------------------------------- utilities
__global__ void zero_kernel(float* __restrict__ p, int n) {
  int i = blockIdx.x * blockDim.x + threadIdx.x;
  if (i < n) p[i] = 0.0f;
}

__global__ void deg_kernel(const int* __restrict__ dst, float* __restrict__ deg, int E) {
  int e = blockIdx.x * blockDim.x + threadIdx.x;
  if (e < E) atomAdd(&deg[dst[e]], 1.0f);
}

// dis = (deg + 1)^-1/2   (+1 accounts for the self-loop; always > 0)
__global__ void dis_kernel(const float* __restrict__ deg, float* __restrict__ dis, int N) {
  int i = blockIdx.x * blockDim.x + threadIdx.x;
  if (i < N) dis[i] = rsqrtf(deg[i] + 1.0f);
}

__global__ void counts_kernel(const int* __restrict__ batch, float* __restrict__ counts, int N) {
  int i = blockIdx.x * blockDim.x + threadIdx.x;
  if (i < N) atomAdd(&counts[batch[i]], 1.0f);
}

// ------------------------------------------------------- layer 0 (K=3) GEMM
// out[n,c] = sum_k x[n, tOff+k] * W0[k,c]  — 3 FMAs/elem, memory bound.
__global__ void lin3_kernel(const float* __restrict__ x, const float* __restrict__ W0,
                            float* __restrict__ out, int N, int tOff) {
  int idx = blockIdx.x * blockDim.x + threadIdx.x;
  if (idx >= N * DCH) return;
  int n = idx / DCH, c = idx - n * DCH;
  const float* xr = x + (size_t)n * (TSTEPS * 3) + tOff;
  out[idx] = fmaf(xr[0], W0[c], fmaf(xr[1], W0[DCH + c], xr[2] * W0[2 * DCH + c]));
}

// ----------------------------------------------------------- K=192 WMMA GEMM
// C[M x 192chunk] = A[M x 192] * B[192 x 192chunk] (B viewed through ldb).
// Block: 128 threads = 4 waves; block tile 64 rows x 64 cols.
// Each wave: 16 rows x 64 cols = 4 accumulators.
// B panel staged in LDS *fragment-major*: group g = (k0/4)*2 + half holds the
// K-row pair {kg, kg+1}; lds[g*128 + 2c + j] = B[kg + j][colBase + c].
// A wave lane then reads each 2-element B fragment as ONE aligned ds_load_b64.
// grid: (ceil(Mtiles/4), DCH/64).
__global__ __launch_bounds__(128)
void gemm_wmma_f32(const float* __restrict__ A, int lda,
                   const float* __restrict__ B, int ldb,
                   float* __restrict__ C, int Mtiles) {
  __shared__ float ldsB[(KDIM / 2) * 128];   // 48 KB panel (permuted)

  const int tid     = threadIdx.x;
  const int colBase = blockIdx.y * 64;

  // Stage: each iteration handles one (K-row-pair, column) cell.
  // Column index is thread-fast -> coalesced global_load_b32 pairs,
  // one contiguous ds_store_b64 per cell.
  for (int i = tid; i < (KDIM / 2) * 64; i += 128) {
    int c = i & 63;
    int g = i >> 6;                        // fragment group 0..95
    int k = (g >> 1) * 4 + (g & 1) * 2;    // first K-row of the pair
    v2f val;
    val.x = B[(size_t)k       * ldb + colBase + c];
    val.y = B[(size_t)(k + 1) * ldb + colBase + c];
    *(v2f*)&ldsB[g * 128 + c * 2] = val;
  }
  __syncthreads();

  const int lane  = tid & 31;
  const int wave  = tid >> 5;
  const int mTile = blockIdx.x * 4 + wave;
  if (mTile >= Mtiles) return;               // no more barriers after this point

  const int half = lane >> 4;                // 0: lanes 0-15, 1: lanes 16-31
  const int l15  = lane & 15;
  // A 16x4 fragment (ISA 7.12.2): lane row = l15, K pair {0,1} or {2,3} by half.
  const float* Ap = A + (size_t)(mTile * 16 + l15) * lda + 2 * half;
  // B fragment base for this lane: group stride 256 floats per k0 step of 4.
  const float* Bp = ldsB + half * 128 + l15 * 2;

  v8f acc0 = {}, acc1 = {}, acc2 = {}, acc3 = {};

#pragma unroll 4
  for (int k0 = 0; k0 < KDIM; k0 += 4) {
    v2f a = *(const v2f*)(Ap + k0);          // one global_load_b64, immediate offset
    const float* bq = Bp + (k0 >> 2) * 256;
    v2f b0 = *(const v2f*)(bq);              // ds_load_b64, even VGPR pair
    v2f b1 = *(const v2f*)(bq + 32);         // +16 cols
    v2f b2 = *(const v2f*)(bq + 64);         // +32 cols
    v2f b3 = *(const v2f*)(bq + 96);         // +48 cols
    acc0 = __builtin_amdgcn_wmma_f32_16x16x4_f32(false, a, false, b0, (short)0, acc0, false, false);
    acc1 = __builtin_amdgcn_wmma_f32_16x16x4_f32(false, a, false, b1, (short)0, acc1, false, false);
    acc2 = __builtin_amdgcn_wmma_f32_16x16x4_f32(false, a, false, b2, (short)0, acc2, false, false);
    acc3 = __builtin_amdgcn_wmma_f32_16x16x4_f32(false, a, false, b3, (short)0, acc3, false, false);
  }

  // D layout: VGPR v -> row = tileRow + v + 8*half, col = colBase + 16*ct + l15.
  const int rBase = mTile * 16 + 8 * half;
  float* Cp = C + (size_t)rBase * DCH + colBase + l15;
#pragma unroll
  for (int v = 0; v < 8; ++v) {
    float* cr = Cp + (size_t)v * DCH;
    cr[0]  = acc0[v];
    cr[16] = acc1[v];
    cr[32] = acc2[v];
    cr[48] = acc3[v];
  }
}

// -------------------------------------------------------------- aggregation
// agg[dst, :] += hw[src, :] * (dis[src]*dis[dst]); self-loops e in [E, E+N).
// Thread handles one edge x 4 channels (float4 gather, channel-fast mapping).
__global__ void agg_kernel(const float* __restrict__ hw,
                           const int* __restrict__ src, const int* __restrict__ dst,
                           const float* __restrict__ dis,
                           float* __restrict__ agg, int E, int N) {
  int tid = blockIdx.x * blockDim.x + threadIdx.x;
  const int CP = DCH / 4;                    // 48 channel packs
  int e = tid / CP;
  int c = (tid - e * CP) * 4;
  if (e >= E + N) return;
  int s, d;
  if (e < E) { s = src[e]; d = dst[e]; }
  else       { s = d = e - E; }
  float en = dis[s] * dis[d];
  const float4 v = *(const float4*)(hw + (size_t)s * DCH + c);
  float* o = agg + (size_t)d * DCH + c;
  atomAdd(o + 0, v.x * en);
  atomAdd(o + 1, v.y * en);
  atomAdd(o + 2, v.z * en);
  atomAdd(o + 3, v.w * en);
}

// ------------------------------------------------------------ norm passes
// A: h = agg + bias; meansum[g,c] += h
__global__ void passA_kernel(const float* __restrict__ agg, const float* __restrict__ bias,
                             const int* __restrict__ batch,
                             float* __restrict__ h, float* __restrict__ meansum, int N) {
  int idx = blockIdx.x * blockDim.x + threadIdx.x;
  if (idx >= N * DCH) return;
  int n = idx / DCH, c = idx - n * DCH;
  float v = agg[idx] + bias[c];
  h[idx] = v;
  atomAdd(&meansum[batch[n] * DCH + c], v);
}

// B: sub = h - ga*mean; varsum[g,c] += sub*sub
__global__ void passB_kernel(const float* __restrict__ h, const float* __restrict__ ga,
                             const int* __restrict__ batch,
                             const float* __restrict__ meansum, const float* __restrict__ counts,
                             float* __restrict__ varsum, int N) {
  int idx = blockIdx.x * blockDim.x + threadIdx.x;
  if (idx >= N * DCH) return;
  int n = idx / DCH, c = idx - n * DCH;
  int g = batch[n];
  float mean = meansum[g * DCH + c] / counts[g];
  float sub  = h[idx] - ga[c] * mean;
  atomAdd(&varsum[g * DCH + c], sub * sub);
}

// C: y = gw*sub*rsqrt(var+eps)+gb; exact GELU; optional residual; strided write
__global__ void passC_kernel(const float* __restrict__ h, const float* __restrict__ gw,
                             const float* __restrict__ gb, const float* __restrict__ ga,
                             const int* __restrict__ batch,
                             const float* __restrict__ meansum, const float* __restrict__ varsum,
                             const float* __restrict__ counts, const float* __restrict__ resid,
                             float* __restrict__ out, int N, int rowStride, int colOff) {
  int idx = blockIdx.x * blockDim.x + threadIdx.x;
  if (idx >= N * DCH) return;
  int n = idx / DCH, c = idx - n * DCH;
  int g = batch[n];
  float cnt  = counts[g];
  float mean = meansum[g * DCH + c] / cnt;
  float var  = varsum[g * DCH + c] / cnt;
  float sub  = h[idx] - ga[c] * mean;
  float y = gw[c] * sub * rsqrtf(var + EPSV) + gb[c];
  y = 0.5f * y * (1.0f + erff(y * 0.70710678118654752f));   // exact-erf GELU
  if (resid) y += resid[idx];
  out[(size_t)n * rowStride + colOff + c] = y;
}

// ---------------------------------------------------------------- launcher
extern "C" void kernel_launch(void* const* d_in, const int* in_sizes, int n_in,
                              void* d_out, int out_size, void* d_ws, size_t ws_size,
                              hipStream_t stream) {
  const float* x     = (const float*)d_in[0];     // [N, T, 3]
  const int*   ei    = (const int*)d_in[1];       // [2, E]
  const int*   batch = (const int*)d_in[2];       // [N]
  const float* W[3]  = {(const float*)d_in[3],  (const float*)d_in[8],  (const float*)d_in[13]};
  const float* B[3]  = {(const float*)d_in[4],  (const float*)d_in[9],  (const float*)d_in[14]};
  const float* GW[3] = {(const float*)d_in[5],  (const float*)d_in[10], (const float*)d_in[15]};
  const float* GB[3] = {(const float*)d_in[6],  (const float*)d_in[11], (const float*)d_in[16]};
  const float* GA[3] = {(const float*)d_in[7],  (const float*)d_in[12], (const float*)d_in[17]};

  const int N = in_sizes[0] / (TSTEPS * 3);
  const int E = in_sizes[1] / 2;
  const int* src = ei;
  const int* dst = ei + E;

  // workspace layout (floats): agg, meansum, varsum kept contiguous for one zero pass
  float* ws = (float*)d_ws;
  size_t off = 0;
  float* deg     = ws + off; off += (size_t)N;
  float* dis     = ws + off; off += (size_t)N;
  float* counts  = ws + off; off += NG;
  float* agg     = ws + off; off += (size_t)N * DCH;
  float* meansum = ws + off; off += (size_t)NG * DCH;
  float* varsum  = ws + off; off += (size_t)NG * DCH;
  float* bufA    = ws + off; off += (size_t)N * DCH;   // layer input h (N x 192)
  float* bufB    = ws + off; off += (size_t)N * DCH;   // gemm out / biased h

  const int TB = 256;
  auto blocks = [](long long n, int tb) { return (unsigned)((n + tb - 1) / tb); };

  // --- one-time precompute (per call; deterministic) ---
  zero_kernel<<<blocks(N, TB), TB, 0, stream>>>(deg, N);
  zero_kernel<<<1, NG, 0, stream>>>(counts, NG);
  deg_kernel<<<blocks(E, TB), TB, 0, stream>>>(dst, deg, E);
  dis_kernel<<<blocks(N, TB), TB, 0, stream>>>(deg, dis, N);
  counts_kernel<<<blocks(N, TB), TB, 0, stream>>>(batch, counts, N);

  const int statCount = N * DCH + 2 * NG * DCH;  // agg + meansum + varsum (contiguous)
  const long long aggThreads = (long long)(E + N) * (DCH / 4);
  const int nd     = N * DCH;
  const int Mtiles = (N + 15) / 16;

  // One layer-chunk pipeline. K3: layer-0 path (tOff = t*3); else K=192 WMMA GEMM.
  auto chunk = [&](bool k3, const float* A, int lda, int tOff,
                   const float* Wp, int ldb,
                   const float* bias, const float* gw, const float* gb, const float* ga,
                   const float* resid, float* out, int rowStride, int colOff) {
    zero_kernel<<<blocks(statCount, TB), TB, 0, stream>>>(agg, statCount);
    if (k3) {
      lin3_kernel<<<blocks(nd, TB), TB, 0, stream>>>(x, Wp, bufB, N, tOff);
    } else {
      dim3 gg((Mtiles + 3) / 4, DCH / 64);
      gemm_wmma_f32<<<gg, 128, 0, stream>>>(A, lda, Wp, ldb, bufB, Mtiles);
    }
    agg_kernel<<<blocks(aggThreads, TB), TB, 0, stream>>>(bufB, src, dst, dis, agg, E, N);
    passA_kernel<<<blocks(nd, TB), TB, 0, stream>>>(agg, bias, batch, bufB, meansum, N);
    passB_kernel<<<blocks(nd, TB), TB, 0, stream>>>(bufB, ga, batch, meansum, counts, varsum, N);
    passC_kernel<<<blocks(nd, TB), TB, 0, stream>>>(bufB, gw, gb, ga, batch, meansum, varsum,
                                                    counts, resid, out, N, rowStride, colOff);
  };

  float* outp = (float*)d_out;  // [N, T, 768]
  for (int t = 0; t < TSTEPS; ++t) {
    // layer 0: x[:,t,:] (N x 3) @ W0 (3 x 192) -> bufA, no residual
    chunk(true, nullptr, 0, t * 3, W[0], DCH,
          B[0], GW[0], GB[0], GA[0], nullptr, bufA, DCH, 0);
    // layer 1: bufA (N x 192) @ W1 (192 x 192) -> bufA, residual = bufA
    chunk(false, bufA, DCH, 0, W[1], DCH,
          B[1], GW[1], GB[1], GA[1], bufA, bufA, DCH, 0);
    // layer 2: bufA @ W2 (192 x 768), 4 column chunks straight into d_out
    for (int j = 0; j < 4; ++j) {
      chunk(false, bufA, DCH, 0, W[2] + j * DCH, 768,
            B[2] + j * DCH, GW[2] + j * DCH, GB[2] + j * DCH, GA[2] + j * DCH,
            nullptr, outp, TSTEPS * 768, t * 768 + j * DCH);
    }
  }
}